// GetModel_46248207843942
// MI455X (gfx1250) — compile-verified
//
#include <hip/hip_runtime.h>
#include <float.h>
#include <math.h>

// ---------------------------------------------------------------------------
// Problem constants (from reference)
// ---------------------------------------------------------------------------
#define BB   8
#define FF   2048
#define NN   10000
#define JJ   10
#define RR   20
#define H1N  200
#define H2N  100
#define MPAD 16          // J padded to WMMA M dimension

typedef __attribute__((ext_vector_type(2))) float v2f;
typedef __attribute__((ext_vector_type(8))) float v8f;

// ---------------------------------------------------------------------------
// Kernel 0: pad conv_w [10,2048] -> Apad [16,2048] (rows 10..15 = 0)
// ---------------------------------------------------------------------------
__global__ void pad_w_kernel(const float* __restrict__ conv_w,
                             float* __restrict__ Apad) {
    int i = blockIdx.x * 256 + threadIdx.x;
    if (i < MPAD * FF) {
        int m = i / FF;
        int f = i % FF;
        Apad[i] = (m < JJ) ? conv_w[m * FF + f] : 0.0f;
    }
}

// ---------------------------------------------------------------------------
// Kernel 1: emb[b,j,n] = sum_f x[b,f,n]*w[j,f] + conv_b[j]
// via V_WMMA_F32_16X16X4_F32.  Each wave: two 16x16 output tiles, K=2048.
// A (16x4 f32): lanes 0-15 hold M=lane, VGPR0=K0,VGPR1=K1; lanes 16-31: K2,K3.
// B (4x16 f32): lanes 0-15 hold N=lane, VGPR0=K0,VGPR1=K1; lanes 16-31: K2,K3.
// ---------------------------------------------------------------------------
__global__ void conv_wmma_kernel(const float* __restrict__ x,
                                 const float* __restrict__ Apad,
                                 const float* __restrict__ conv_b,
                                 float* __restrict__ emb) {
    const int b    = blockIdx.y;
    const int tid  = threadIdx.x;
    const int wave = tid >> 5;
    const int lane = tid & 31;
    const int half = lane >> 4;     // 0: K0/K1 rows, 1: K2/K3 rows
    const int lcol = lane & 15;     // N column / M row within half

    // two consecutive 16-column tiles per wave
    const int tile0 = (blockIdx.x * 8 + wave) * 2;
    const int nb0   = tile0 * 16;
    const int nb1   = nb0 + 16;
    const bool v0   = (nb0 < NN);           // wave-uniform
    const bool v1   = (nb1 < NN);           // wave-uniform
    if (!v0) return;                        // whole wave exits; EXEC stays full
    const int nb1c  = v1 ? nb1 : nb0;       // safe address for dead tile

    const float* __restrict__ xb = x + (size_t)b * FF * NN;
    // A address for this lane: row m = lcol, K offset = 2*half
    const float* __restrict__ arow = Apad + (size_t)lcol * FF + 2 * half;

    v8f c0 = {};
    v8f c1 = {};

#pragma unroll 4
    for (int k = 0; k < FF / 4; ++k) {
        const int f0 = 4 * k;
        // A fragment: {w[m][f0+2h], w[m][f0+2h+1]} -> one global_load_b64
        v2f a = *(const v2f*)(arow + f0);

        const int fb = f0 + 2 * half;
        const float* p0 = xb + (size_t)fb * NN;
        const float* p1 = xb + (size_t)(fb + 1) * NN;

        v2f bv0 = { p0[nb0 + lcol], p1[nb0 + lcol] };
        v2f bv1 = { p0[nb1c + lcol], p1[nb1c + lcol] };

        c0 = __builtin_amdgcn_wmma_f32_16x16x4_f32(
                 false, a, false, bv0, (short)0, c0, false, false);
        c1 = __builtin_amdgcn_wmma_f32_16x16x4_f32(
                 false, a, false, bv1, (short)0, c1, false, false);
    }

    // C/D layout: lane group selects N col; VGPR v -> M = v + half*8
    const int col0 = nb0 + lcol;
    const int col1 = nb1 + lcol;
#pragma unroll
    for (int v = 0; v < 8; ++v) {
        const int m = v + half * 8;
        if (m < JJ) {
            const float bias = conv_b[m];
            float* __restrict__ row = emb + ((size_t)b * JJ + m) * NN;
            row[col0] = c0[v] + bias;
            if (v1) row[col1] = c1[v] + bias;
        }
    }
}

// ---------------------------------------------------------------------------
// Kernel 2: per (b,j) row of emb, write mm[b,j,0..19]  = top-20 descending,
//                                  mm[b,j,20..39] = bottom-20 descending.
// One block per row; row cached in LDS (40KB of the 320KB WGP pool).
// ---------------------------------------------------------------------------
__global__ void minmax_kernel(const float* __restrict__ emb,
                              float* __restrict__ mm) {
    __shared__ float vals[NN];
    __shared__ float rv[256];
    __shared__ int   ri[256];
    __shared__ float topval[RR];
    __shared__ int   topidx[RR];

    const int row = blockIdx.x;              // b*JJ + j
    const int tid = threadIdx.x;
    const float* __restrict__ src = emb + (size_t)row * NN;
    float* __restrict__ out = mm + (size_t)row * (2 * RR);

    for (int i = tid; i < NN; i += 256) vals[i] = src[i];
    __syncthreads();

    // ---- top-20 (descending) ----
    for (int r = 0; r < RR; ++r) {
        float lv = -FLT_MAX; int li = 0;
        for (int i = tid; i < NN; i += 256) {
            float v = vals[i];
            if (v > lv) { lv = v; li = i; }
        }
        rv[tid] = lv; ri[tid] = li;
        __syncthreads();
        for (int s = 128; s > 0; s >>= 1) {
            if (tid < s) {
                float vo = rv[tid + s];
                if (vo > rv[tid] || (vo == rv[tid] && ri[tid + s] < ri[tid])) {
                    rv[tid] = vo; ri[tid] = ri[tid + s];
                }
            }
            __syncthreads();
        }
        if (tid == 0) {
            out[r]    = rv[0];
            topval[r] = rv[0];
            topidx[r] = ri[0];
            vals[ri[0]] = -FLT_MAX;          // remove from consideration
        }
        __syncthreads();
    }

    // restore removed maxima before the min pass
    if (tid < RR) vals[topidx[tid]] = topval[tid];
    __syncthreads();

    // ---- bottom-20 (extract ascending, store descending) ----
    for (int r = 0; r < RR; ++r) {
        float lv = FLT_MAX; int li = 0;
        for (int i = tid; i < NN; i += 256) {
            float v = vals[i];
            if (v < lv) { lv = v; li = i; }
        }
        rv[tid] = lv; ri[tid] = li;
        __syncthreads();
        for (int s = 128; s > 0; s >>= 1) {
            if (tid < s) {
                float vo = rv[tid + s];
                if (vo < rv[tid] || (vo == rv[tid] && ri[tid + s] < ri[tid])) {
                    rv[tid] = vo; ri[tid] = ri[tid + s];
                }
            }
            __syncthreads();
        }
        if (tid == 0) {
            out[2 * RR - 1 - r] = rv[0];     // r-th smallest -> position 39-r
            vals[ri[0]] = FLT_MAX;
        }
        __syncthreads();
    }
}

// ---------------------------------------------------------------------------
// Kernel 3: tiny MLP, one block.  h0[8,400] -> sigmoid(w1,b1) -> sigmoid(w2,b2)
// -> w3,b3 -> out[8,2].
// ---------------------------------------------------------------------------
__device__ __forceinline__ float sigmoidf(float v) {
    return 1.0f / (1.0f + __expf(-v));
}

__global__ void mlp_kernel(const float* __restrict__ mm,
                           const float* __restrict__ w1, const float* __restrict__ b1,
                           const float* __restrict__ w2, const float* __restrict__ b2,
                           const float* __restrict__ w3, const float* __restrict__ b3,
                           float* __restrict__ out) {
    __shared__ float h0[BB * 2 * RR * JJ / JJ * JJ]; // 8*400
    __shared__ float h1[BB * H1N];                   // 8*200
    __shared__ float h2[BB * H2N];                   // 8*100
    const int tid = threadIdx.x;
    const int IN1 = 2 * RR * JJ;                     // 400

    for (int i = tid; i < BB * IN1; i += 256) h0[i] = mm[i];
    __syncthreads();

    for (int idx = tid; idx < BB * H1N; idx += 256) {
        const int b = idx / H1N, o = idx % H1N;
        const float* __restrict__ h = h0 + b * IN1;
        float acc = b1[o];
        for (int k = 0; k < IN1; ++k) acc += h[k] * w1[k * H1N + o];
        h1[idx] = sigmoidf(acc);
    }
    __syncthreads();

    for (int idx = tid; idx < BB * H2N; idx += 256) {
        const int b = idx / H2N, o = idx % H2N;
        const float* __restrict__ h = h1 + b * H1N;
        float acc = b2[o];
        for (int k = 0; k < H1N; ++k) acc += h[k] * w2[k * H2N + o];
        h2[idx] = sigmoidf(acc);
    }
    __syncthreads();

    for (int idx = tid; idx < BB * 2; idx += 256) {
        const int b = idx / 2, o = idx % 2;
        const float* __restrict__ h = h2 + b * H2N;
        float acc = b3[o];
        for (int k = 0; k < H2N; ++k) acc += h[k] * w3[k * 2 + o];
        out[idx] = acc;
    }
}

// ---------------------------------------------------------------------------
// Launch
// ---------------------------------------------------------------------------
extern "C" void kernel_launch(void* const* d_in, const int* in_sizes, int n_in,
                              void* d_out, int out_size, void* d_ws, size_t ws_size,
                              hipStream_t stream) {
    const float* x      = (const float*)d_in[0];
    const float* conv_w = (const float*)d_in[1];
    const float* conv_b = (const float*)d_in[2];
    const float* w1     = (const float*)d_in[3];
    const float* b1     = (const float*)d_in[4];
    const float* w2     = (const float*)d_in[5];
    const float* b2     = (const float*)d_in[6];
    const float* w3     = (const float*)d_in[7];
    const float* b3     = (const float*)d_in[8];

    float* wsf  = (float*)d_ws;
    float* Apad = wsf;                                   // 16*2048       = 32768
    float* emb  = wsf + MPAD * FF;                       // 8*10*10000    = 800000
    float* mm   = emb + (size_t)BB * JJ * NN;            // 8*10*40       = 3200

    // 0) pad A to 16 rows
    pad_w_kernel<<<(MPAD * FF + 255) / 256, 256, 0, stream>>>(conv_w, Apad);

    // 1) conv via WMMA: 625 tiles of 16 cols per batch, 2 tiles/wave,
    //    8 waves/block -> 40 blocks in x, B in y.
    conv_wmma_kernel<<<dim3(40, BB), 256, 0, stream>>>(x, Apad, conv_b, emb);

    // 2) top/bottom-20 per (b,j)
    minmax_kernel<<<BB * JJ, 256, 0, stream>>>(emb, mm);

    // 3) MLP head
    mlp_kernel<<<1, 256, 0, stream>>>(mm, w1, b1, w2, b2, w3, b3, (float*)d_out);
}